// LogicConvUnfold_6897717477608
// MI455X (gfx1250) — compile-verified
//
#include <hip/hip_runtime.h>

// ---------------------------------------------------------------------------
// LogicConvUnfold for MI455X (gfx1250).
// out[b,k,oh,ow] = C0[k] + Ca[k]*a + Cb[k]*b + Cab[k]*a*b
//   a = x[b, ca[k], oh+ha[k], ow+wa[k]],  b = x[b, cb[k], oh+hb[k], ow+wb[k]]
// Store-bandwidth bound (~65MB out @ 23.3 TB/s). x (4MB) stays in L2.
// ---------------------------------------------------------------------------

constexpr int Bc = 8, Cc = 8, Hc = 128, Wc = 128;
constexpr int Kc = 128;
constexpr int OHc = 126, OWc = 126;
constexpr int ROWS = 14;   // output rows per band
constexpr int NB   = 9;    // 14*9 == 126

struct KParam {
  float c0, ca, cb, cab;   // field order matches WMMA N columns 0..3
  int baseA, wa, baseB, wb;
};

typedef __attribute__((ext_vector_type(2))) float v2f;
typedef __attribute__((ext_vector_type(8))) float v8f;
typedef __attribute__((ext_vector_type(4))) int   v4i;
typedef __attribute__((address_space(1))) v4i global_v4i;
typedef __attribute__((address_space(3))) v4i shared_v4i;

#if defined(__HIP_DEVICE_COMPILE__)
#if __has_builtin(__builtin_amdgcn_global_load_async_to_lds_b128)
#define HAVE_ASYNC_COPY 1
#endif
#if __has_builtin(__builtin_amdgcn_wmma_f32_16x16x4_f32)
#define HAVE_WMMA_F32X4 1
#endif
#endif

__device__ __forceinline__ void g2lds_b128(const float* g, float* l) {
#if defined(HAVE_ASYNC_COPY)
  // Async DMA global -> LDS, 16B per lane; tracked with ASYNCcnt.
  __builtin_amdgcn_global_load_async_to_lds_b128(
      (global_v4i*)(void*)const_cast<float*>(g),
      (shared_v4i*)(void*)l,
      0, 0);
#else
  *reinterpret_cast<float4*>(l) = *reinterpret_cast<const float4*>(g);
#endif
}

__device__ __forceinline__ void wait_async_copies() {
#if defined(HAVE_ASYNC_COPY)
#if __has_builtin(__builtin_amdgcn_s_wait_asynccnt)
  __builtin_amdgcn_s_wait_asynccnt(0);
#else
  asm volatile("s_wait_asynccnt 0" ::: "memory");
#endif
#endif
}

// T[16][4]: columns are the 16->4 linear maps for (C0, Ca, Cb, Cab).
__device__ __constant__ float Tmat[16][4] = {
  {0, 0, 0,  0}, {0, 0, 0,  1}, {0, 1, 0, -1}, {0, 1, 0,  0},
  {0, 0, 1, -1}, {0, 0, 1,  0}, {0, 1, 1, -2}, {0, 1, 1, -1},
  {1,-1,-1,  1}, {1,-1,-1,  2}, {1, 0,-1,  0}, {1, 0,-1,  1},
  {1,-1, 0,  0}, {1,-1, 0,  1}, {1, 0, 0, -1}, {1, 0, 0,  0}
};

// One wave (32 lanes). coeff(128x4) = weights(128x16) @ T(16x4) via
// 8 row-blocks x 4 chained v_wmma_f32_16x16x4_f32, plus per-k gather offsets.
__global__ __launch_bounds__(32) void precompute_kernel(
    const float* __restrict__ w, const int* __restrict__ pa,
    const int* __restrict__ pb, KParam* __restrict__ prm) {
  const int lane = threadIdx.x;       // 0..31, full EXEC for WMMA
  const int half = lane >> 4;         // 0: lanes 0-15, 1: lanes 16-31
  const int n    = lane & 15;

#if defined(HAVE_WMMA_F32X4)
  for (int blk = 0; blk < 8; ++blk) {
    v8f acc = {};
#pragma unroll
    for (int j = 0; j < 4; ++j) {
      // A = weights[blk*16 .. +15][4j .. 4j+3]  (16x4 fp32, 2 VGPRs/lane)
      const int row = blk * 16 + n;
      v2f a;
      a.x = w[row * 16 + 4 * j + 2 * half + 0];
      a.y = w[row * 16 + 4 * j + 2 * half + 1];
      // B = T[4j .. 4j+3][0..15] (4x16, cols >=4 zero-padded)
      const int kr = 4 * j + 2 * half;
      v2f bm;
      bm.x = (n < 4) ? Tmat[kr + 0][n] : 0.0f;
      bm.y = (n < 4) ? Tmat[kr + 1][n] : 0.0f;
      acc = __builtin_amdgcn_wmma_f32_16x16x4_f32(
          /*neg_a=*/false, a, /*neg_b=*/false, bm,
          /*c_mod=*/(short)0, acc, /*reuse_a=*/false, /*reuse_b=*/false);
    }
    if (n < 4) {
#pragma unroll
      for (int i = 0; i < 8; ++i) {
        const int m = blk * 16 + i + 8 * half;   // D: VGPR i -> M=i / i+8
        ((float*)&prm[m])[n] = acc[i];           // n in {0,1,2,3} -> c0..cab
      }
    }
  }
#else
  for (int k = lane; k < Kc; k += 32) {
    const float* wr = w + k * 16;
    prm[k].c0  = wr[8]+wr[9]+wr[10]+wr[11]+wr[12]+wr[13]+wr[14]+wr[15];
    prm[k].ca  = wr[2]+wr[3]+wr[6]+wr[7]-wr[8]-wr[9]-wr[12]-wr[13];
    prm[k].cb  = wr[4]+wr[5]+wr[6]+wr[7]-wr[8]-wr[9]-wr[10]-wr[11];
    prm[k].cab = wr[1]-wr[2]-wr[4]-2.0f*wr[6]-wr[7]+wr[8]+2.0f*wr[9]
               + wr[11]+wr[13]-wr[14];
  }
#endif

  for (int k = lane; k < Kc; k += 32) {
    const int ha = pa[3 * k + 0], wa = pa[3 * k + 1], ca = pa[3 * k + 2];
    const int hb = pb[3 * k + 0], wb = pb[3 * k + 1], cb = pb[3 * k + 2];
    prm[k].baseA = ca * Hc * Wc + ha * Wc;
    prm[k].wa    = wa;
    prm[k].baseB = cb * Hc * Wc + hb * Wc;
    prm[k].wb    = wb;
  }
}

// One block = one 14-row band of one (b,k) output plane.
// Stage the two needed 14x128 fp32 windows into LDS via async DMA, then
// stream non-temporal stores of the band.
__global__ __launch_bounds__(256) void logic_conv_kernel(
    const float* __restrict__ x, const KParam* __restrict__ prm,
    float* __restrict__ out) {
  __shared__ __align__(16) float ldsA[ROWS * Wc];
  __shared__ __align__(16) float ldsB[ROWS * Wc];

  const int bid  = blockIdx.x;
  const int band = bid % NB;
  const int bk   = bid / NB;
  const int k    = bk % Kc;
  const int b    = bk / Kc;
  const KParam p = prm[k];          // uniform -> scalar loads

  const int oh0 = band * ROWS;
  const float* xb   = x + b * (Cc * Hc * Wc);
  const float* srcA = xb + p.baseA + oh0 * Wc;   // 16B-aligned (mult. of 128 floats)
  const float* srcB = xb + p.baseB + oh0 * Wc;

  const int tid = threadIdx.x;
  constexpr int CHUNKS = ROWS * Wc / 4;          // 448 x 16B per plane
  for (int i = tid; i < CHUNKS; i += 256) {
    g2lds_b128(srcA + i * 4, &ldsA[i * 4]);
    g2lds_b128(srcB + i * 4, &ldsB[i * 4]);
  }
  wait_async_copies();
  __syncthreads();

  const int col = tid & (Wc - 1);                // 0..127
  const int r0  = tid >> 7;                      // 0..1
  float* dst = out + ((b * Kc + k) * OHc + oh0) * OWc;

  if (col < OWc) {
#pragma unroll
    for (int r = r0; r < ROWS; r += 2) {
      const float av = ldsA[r * Wc + col + p.wa];
      const float bv = ldsB[r * Wc + col + p.wb];
      const float ab = av * bv;
      const float o =
          fmaf(p.cab, ab, fmaf(p.cb, bv, fmaf(p.ca, av, p.c0)));
      __builtin_nontemporal_store(o, dst + r * OWc + col);
    }
  }
}

extern "C" void kernel_launch(void* const* d_in, const int* in_sizes, int n_in,
                              void* d_out, int out_size, void* d_ws,
                              size_t ws_size, hipStream_t stream) {
  const float* x       = (const float*)d_in[0];
  const float* weights = (const float*)d_in[1];
  const int*   pairs_a = (const int*)d_in[2];
  const int*   pairs_b = (const int*)d_in[3];
  float*       out     = (float*)d_out;
  KParam*      prm     = (KParam*)d_ws;   // 128 * 32 B = 4 KB scratch

  precompute_kernel<<<1, 32, 0, stream>>>(weights, pairs_a, pairs_b, prm);
  logic_conv_kernel<<<dim3(Bc * Kc * NB), dim3(256), 0, stream>>>(x, prm, out);
}